// BinaryGraphClassifier_23201413333150
// MI455X (gfx1250) — compile-verified
//
#include <hip/hip_runtime.h>
#include <hip/hip_bf16.h>

typedef __attribute__((ext_vector_type(2))) float v2f;
typedef __attribute__((ext_vector_type(8))) float v8f;

#define DH 128
#define LDA 132   // LDS row stride (floats): bank-conflict-free, keeps 16B alignment

// ---------------- utility kernels ----------------

__global__ __launch_bounds__(256) void zero4_kernel(float4* __restrict__ p, long n4) {
    long i = (long)blockIdx.x * blockDim.x + threadIdx.x;
    long stride = (long)gridDim.x * blockDim.x;
    float4 z = make_float4(0.f, 0.f, 0.f, 0.f);
    for (; i < n4; i += stride) p[i] = z;
}

__global__ __launch_bounds__(256) void zero1_kernel(float* __restrict__ p, int n) {
    int i = blockIdx.x * blockDim.x + threadIdx.x;
    if (i < n) p[i] = 0.f;
}

// degree: one thread per edge, atomic +1 on dst
__global__ __launch_bounds__(256) void degree_kernel(const int* __restrict__ dst,
                                                     float* __restrict__ deg, int E) {
    int e = blockIdx.x * blockDim.x + threadIdx.x;
    if (e < E) unsafeAtomicAdd(&deg[dst[e]], 1.0f);
}

// norm[i] = clip(deg,1)^-0.5   (in place on deg buffer)
__global__ __launch_bounds__(256) void norm_kernel(float* __restrict__ d, int n) {
    int i = blockIdx.x * blockDim.x + threadIdx.x;
    if (i < n) d[i] = rsqrtf(fmaxf(d[i], 1.0f));
}

// ---------------- graph propagation ----------------
// one wave32 per edge; lane handles 4 consecutive features (float4 gather, 4 f32 atomics)
__global__ __launch_bounds__(256) void scatter_edges_kernel(
    const float* __restrict__ h, const float* __restrict__ norm,
    const int* __restrict__ src, const int* __restrict__ dst,
    float* __restrict__ out, int E) {
    long gid = (long)blockIdx.x * blockDim.x + threadIdx.x;
    int e = (int)(gid >> 5);
    int lane = (int)(gid & 31);
    if (e >= E) return;
    int s = src[e];
    int d = dst[e];
    float ns = norm[s];
    const float4* hv = (const float4*)(h + (size_t)s * DH);
    float4 v = hv[lane];
    float* op = out + (size_t)d * DH + lane * 4;
    unsafeAtomicAdd(op + 0, v.x * ns);
    unsafeAtomicAdd(op + 1, v.y * ns);
    unsafeAtomicAdd(op + 2, v.z * ns);
    unsafeAtomicAdd(op + 3, v.w * ns);
}

// out[r][c] *= norm[r]
__global__ __launch_bounds__(256) void scale_rows_kernel(float* __restrict__ h,
                                                         const float* __restrict__ norm,
                                                         long total) {
    long i = (long)blockIdx.x * blockDim.x + threadIdx.x;
    long stride = (long)gridDim.x * blockDim.x;
    for (; i < total; i += stride) h[i] *= norm[i >> 7];
}

// ---------------- dense layer: Y = relu(H @ W + b), W is DHxDH f32, via WMMA f32 16x16x4 ----
// block = 256 threads (8 waves). Block handles 16 rows; wave w owns cols [16w,16w+16).
__global__ __launch_bounds__(256) void gemm_relu_wmma_kernel(
    const float* __restrict__ H, const float* __restrict__ W,
    const float* __restrict__ bias, float* __restrict__ Y) {
    __shared__ float lds[16 * LDA];
    const int tid = threadIdx.x;
    const long rowbase = (long)blockIdx.x * 16;

    // stage 16x128 f32 A-strip into LDS (512 float4, 2 per thread)
    const float4* src4 = (const float4*)(H + (size_t)rowbase * DH);
    #pragma unroll
    for (int i = tid; i < 512; i += 256) {
        int r  = i >> 5;      // row 0..15
        int c4 = i & 31;      // float4 column chunk
        float4 v = src4[r * 32 + c4];
        float* dp = &lds[r * LDA + c4 * 4];
        dp[0] = v.x; dp[1] = v.y; dp[2] = v.z; dp[3] = v.w;
    }
    __syncthreads();

    const int wave = tid >> 5;
    const int lane = tid & 31;
    const int half = lane >> 4;   // 0: lanes 0-15, 1: lanes 16-31
    const int l16  = lane & 15;
    const int col  = wave * 16 + l16;
    const int koff = half * 2;    // upper lane half carries K+2 / K+3
    const float* arow = &lds[l16 * LDA];

    v8f acc = {};
    #pragma unroll
    for (int k = 0; k < DH; k += 4) {
        v2f a, b;
        a.x = arow[k + koff];
        a.y = arow[k + koff + 1];
        b.x = W[(size_t)(k + koff) * DH + col];
        b.y = W[(size_t)(k + koff + 1) * DH + col];
        acc = __builtin_amdgcn_wmma_f32_16x16x4_f32(
            /*neg_a=*/false, a, /*neg_b=*/false, b,
            /*c_mod=*/(short)0, acc, /*reuse_a=*/false, /*reuse_b=*/false);
    }

    const float bv = bias[col];
    // C/D layout: VGPR j -> M = j + 8*half, N = l16
    #pragma unroll
    for (int j = 0; j < 8; ++j) {
        long r = rowbase + j + half * 8;
        Y[(size_t)r * DH + col] = fmaxf(acc[j] + bv, 0.f);
    }
}

// scalar tail for rows not covered by 16-row tiles (only launched if N % 16 != 0)
__global__ __launch_bounds__(128) void gemm_relu_tail_kernel(
    const float* __restrict__ H, const float* __restrict__ W,
    const float* __restrict__ bias, float* __restrict__ Y, int row0, int nrows) {
    int r = row0 + blockIdx.x;
    if (r >= row0 + nrows) return;
    int j = threadIdx.x;
    float s = bias[j];
    for (int k = 0; k < DH; ++k) s += H[(size_t)r * DH + k] * W[(size_t)k * DH + j];
    Y[(size_t)r * DH + j] = fmaxf(s, 0.f);
}

// column sums: block = 128 threads (one per feature), block covers 256 rows
__global__ __launch_bounds__(128) void colsum_kernel(const float* __restrict__ h,
                                                     float* __restrict__ hg, int N) {
    int col = threadIdx.x;
    int r0 = blockIdx.x * 256;
    int rend = min(r0 + 256, N);
    float s = 0.f;
    for (int r = r0; r < rend; ++r) s += h[(size_t)r * DH + col];
    unsafeAtomicAdd(&hg[col], s);
}

// final head: hg/N -> relu(@Wf1+bf1) -> relu(@Wf2+bf2) -> sigmoid
__global__ __launch_bounds__(128) void final_mlp_kernel(
    const float* __restrict__ hg, const float* __restrict__ Wf1,
    const float* __restrict__ bf1, const float* __restrict__ Wf2,
    const float* __restrict__ bf2, float* __restrict__ out, float invN) {
    __shared__ float sh[DH];
    __shared__ float red[DH];
    int j = threadIdx.x;
    sh[j] = hg[j] * invN;
    __syncthreads();
    float s = bf1[j];
    for (int i = 0; i < DH; ++i) s += sh[i] * Wf1[(size_t)i * DH + j];
    s = fmaxf(s, 0.f);
    red[j] = s * Wf2[j];
    __syncthreads();
    for (int off = 64; off > 0; off >>= 1) {
        if (j < off) red[j] += red[j + off];
        __syncthreads();
    }
    if (j == 0) {
        float o = fmaxf(red[0] + bf2[0], 0.f);
        out[0] = 1.f / (1.f + expf(-o));
    }
}

// ---------------- host orchestration ----------------

extern "C" void kernel_launch(void* const* d_in, const int* in_sizes, int n_in,
                              void* d_out, int out_size, void* d_ws, size_t ws_size,
                              hipStream_t stream) {
    const float* x   = (const float*)d_in[0];
    const int*   src = (const int*)d_in[1];
    const int*   dst = (const int*)d_in[2];
    const float* W1  = (const float*)d_in[3];
    const float* b1  = (const float*)d_in[4];
    const float* W2  = (const float*)d_in[5];
    const float* b2  = (const float*)d_in[6];
    const float* Wf1 = (const float*)d_in[7];
    const float* bf1 = (const float*)d_in[8];
    const float* Wf2 = (const float*)d_in[9];
    const float* bf2 = (const float*)d_in[10];

    const int N = in_sizes[0] / DH;
    const int E = in_sizes[1];

    // workspace layout (floats): norm[Npad] | bufA[N*DH] | bufB[N*DH] | hg[128]
    float* norm = (float*)d_ws;
    size_t npad = ((size_t)N + 1023) & ~(size_t)1023;
    float* bufA = norm + npad;
    float* bufB = bufA + (size_t)N * DH;
    float* hg   = bufB + (size_t)N * DH;

    const long totalF  = (long)N * DH;
    const int  gridN   = (N + 255) / 256;
    const int  gridE   = (E + 255) / 256;
    const int  gridEW  = (int)(((long)E * 32 + 255) / 256);  // one wave per edge
    const int  gridZ4  = (int)((totalF / 4 + 255) / 256);
    const int  gridSc  = (int)((totalF + 255) / 256);

    // degree -> norm
    zero1_kernel<<<gridN, 256, 0, stream>>>(norm, N);
    degree_kernel<<<gridE, 256, 0, stream>>>(dst, norm, E);
    norm_kernel<<<gridN, 256, 0, stream>>>(norm, N);

    auto hop = [&](const float* cur, float* outb) {
        zero4_kernel<<<gridZ4, 256, 0, stream>>>((float4*)outb, totalF / 4);
        scatter_edges_kernel<<<gridEW, 256, 0, stream>>>(cur, norm, src, dst, outb, E);
        scale_rows_kernel<<<gridSc, 256, 0, stream>>>(outb, norm, totalF);
    };

    auto dense = [&](const float* Hin, const float* W, const float* b, float* Yout) {
        int full = N / 16;
        int rem  = N - full * 16;
        if (full > 0)
            gemm_relu_wmma_kernel<<<full, 256, 0, stream>>>(Hin, W, b, Yout);
        if (rem > 0)
            gemm_relu_tail_kernel<<<rem, 128, 0, stream>>>(Hin, W, b, Yout, full * 16, rem);
    };

    // propagate x three hops -> bufA
    hop(x, bufA);
    hop(bufA, bufB);
    hop(bufB, bufA);
    // h = relu(h @ W1 + b1) -> bufB
    dense(bufA, W1, b1, bufB);
    // propagate three hops -> bufA
    hop(bufB, bufA);
    hop(bufA, bufB);
    hop(bufB, bufA);
    // h = relu(h @ W2 + b2) -> bufB
    dense(bufA, W2, b2, bufB);

    // mean over nodes
    zero1_kernel<<<1, 128, 0, stream>>>(hg, DH);
    colsum_kernel<<<(N + 255) / 256, 128, 0, stream>>>(bufB, hg, N);

    // head
    final_mlp_kernel<<<1, 128, 0, stream>>>(hg, Wf1, bf1, Wf2, bf2,
                                            (float*)d_out, 1.0f / (float)N);
}